// TemporalFlattenMSDeformAttn_29841432773220
// MI455X (gfx1250) — compile-verified
//
#include <hip/hip_runtime.h>
#include <hip/hip_bf16.h>
#include <math.h>

// ---------------------------------------------------------------------------
// Problem constants (fixed by the reference module)
// ---------------------------------------------------------------------------
#define N_HEADS   8
#define N_LEVELS  4
#define D_MODEL   256
#define LEN_Q     8160          // == LEN_IN ; 510 tiles of 16
#define KDIM      256
#define MTILES    (LEN_Q / 16)  // 510

// per-level flattened-temporal grids: height = T*H, width = W
__constant__ const int c_Hs[4]    = {128, 64, 32, 16};
__constant__ const int c_Ws[4]    = { 48, 24, 12,  6};
__constant__ const int c_start[4] = {  0, 6144, 7680, 8064};

typedef __attribute__((ext_vector_type(2))) float v2f;
typedef __attribute__((ext_vector_type(8))) float v8f;
typedef __attribute__((ext_vector_type(4))) int   v4i;

#if __has_builtin(__builtin_amdgcn_global_load_async_to_lds_b128) && \
    __has_builtin(__builtin_amdgcn_s_wait_asynccnt)
#define HAVE_ASYNC_LDS 1
typedef __attribute__((address_space(1))) v4i gv4i;   // global (AS1)
typedef __attribute__((address_space(3))) v4i lv4i;   // LDS    (AS3)
#endif

// ---------------------------------------------------------------------------
// FP32 WMMA GEMM:  C[M,N] = A[M,K] * W[K,N] + bias[N]
//   Block = 256 threads (8 waves). Block output tile = 128 rows x 64 cols.
//   The W panel W[0:256, nbase:nbase+64] (64 KB) is staged into LDS once per
//   block (GLOBAL_LOAD_ASYNC_TO_LDS_B128 when available), then each wave
//   computes one 16-row M-tile across 4 N-tiles: per k-step 1 global b64 (A)
//   + 8 ds_load_b32 (B) feed 4 x V_WMMA_F32_16X16X4_F32 accumulators.
//
// Operand layouts (CDNA5 ISA 7.12.2):
//   A (16x4 f32):  lanes 0-15 -> M=lane, VGPR0/1 = K0/K1; lanes 16-31 K0+2/+3
//   B (4x16 f32):  lanes index N; VGPRs index K with the same half-split
//   C/D (16x16):   VGPR v: lanes 0-15 -> M=v, lanes 16-31 -> M=v+8; N=lane&15
// ---------------------------------------------------------------------------
__global__ __launch_bounds__(256)
void gemm_f32_wmma(const float* __restrict__ A,
                   const float* __restrict__ W,
                   const float* __restrict__ bias,
                   float* __restrict__ C,
                   int N)
{
    __shared__ float lw[KDIM * 64];            // 64 KB staged weight panel

    const int tid   = threadIdx.x;
    const int nbase = blockIdx.x * 64;

    // ---- stage W[0:256, nbase:nbase+64] into LDS -------------------------
    // 4096 float4s, 16 per thread. Row k = i>>4, col4 = i&15.
    #pragma unroll
    for (int j = 0; j < 16; ++j) {
        const int i = j * 256 + tid;
        const int k = i >> 4, c4 = i & 15;
        const float* src = W + (size_t)k * N + nbase + c4 * 4;
#if HAVE_ASYNC_LDS
        __builtin_amdgcn_global_load_async_to_lds_b128(
            (gv4i*)src, (lv4i*)&lw[i * 4], 0, 0);
#else
        *(float4*)&lw[i * 4] = *(const float4*)src;
#endif
    }
#if HAVE_ASYNC_LDS
    __builtin_amdgcn_s_wait_asynccnt(0);
#endif
    __syncthreads();

    // ---- compute ---------------------------------------------------------
    const int wave  = tid >> 5;
    const int lane  = tid & 31;
    const int mtile = blockIdx.y * 8 + wave;
    if (mtile >= MTILES) return;               // wave-uniform: EXEC all-ones

    const int half = lane >> 4;                // K sub-split (0/1)
    const int l16  = lane & 15;
    const int arow = mtile * 16 + l16;

    const float* Ap = A + (size_t)arow * KDIM + half * 2;
    const float* Bp = lw + (size_t)(half * 2) * 64 + l16;   // + nt*16 + k*64

    v8f acc0 = {}, acc1 = {}, acc2 = {}, acc3 = {};
    #pragma unroll 4
    for (int k0 = 0; k0 < KDIM; k0 += 4) {
        v2f a;
        a.x = Ap[k0];
        a.y = Ap[k0 + 1];
        const float* b0 = Bp + (size_t)k0 * 64;
        v2f b;
        b.x = b0[0];       b.y = b0[64];
        acc0 = __builtin_amdgcn_wmma_f32_16x16x4_f32(false, a, false, b, (short)0, acc0, false, false);
        b.x = b0[16];      b.y = b0[16 + 64];
        acc1 = __builtin_amdgcn_wmma_f32_16x16x4_f32(false, a, false, b, (short)0, acc1, false, false);
        b.x = b0[32];      b.y = b0[32 + 64];
        acc2 = __builtin_amdgcn_wmma_f32_16x16x4_f32(false, a, false, b, (short)0, acc2, false, false);
        b.x = b0[48];      b.y = b0[48 + 64];
        acc3 = __builtin_amdgcn_wmma_f32_16x16x4_f32(false, a, false, b, (short)0, acc3, false, false);
    }

    // ---- writeback: row = mtile*16 + half*8 + v, col = nbase + nt*16 + l16
    float* Crow = C + (size_t)(mtile * 16 + half * 8) * N + nbase + l16;
    v8f* accs[4] = {&acc0, &acc1, &acc2, &acc3};
    #pragma unroll
    for (int nt = 0; nt < 4; ++nt) {
        const float bval = bias[nbase + nt * 16 + l16];
        v8f av = *accs[nt];
        #pragma unroll
        for (int v = 0; v < 8; ++v)
            Crow[(size_t)v * N + nt * 16] = av[v] + bval;
    }
}

// ---------------------------------------------------------------------------
// Deformable sampling kernel.
//   One wave32 per query. lane = head*4 + dchunk ; each lane owns 8 dims of
//   one head.  Per head: 32-logit joint softmax (current + temporal points),
//   then 4 levels x 8 points of bilinear gathers (two b128 per corner).
// ---------------------------------------------------------------------------
__global__ __launch_bounds__(256)
void msda_sample(const float* __restrict__ ref,     // [Q][4][2]
                 const float* __restrict__ toff,    // [Q][4][2][2]
                 const float* __restrict__ value,   // [Q][256]
                 const float* __restrict__ coff,    // [Q][256] (8,4,4,2)
                 const float* __restrict__ cattn,   // [Q][128] (8,16)
                 const float* __restrict__ tsoff,   // [Q][256] (8,4,2,2,2)
                 const float* __restrict__ tattn,   // [Q][128] (8,16)
                 float* __restrict__ out)           // [Q][256]
{
    const int q    = blockIdx.x * 8 + (threadIdx.x >> 5);
    const int lane = threadIdx.x & 31;
    const int head = lane >> 2;
    const int d0   = (lane & 3) * 8;

    // ---- joint softmax over 32 logits: [c_attn(16) | t_attn(16)] ----
    float lg[32];
    const float* ca = cattn + (size_t)q * 128 + head * 16;
    const float* ta = tattn + (size_t)q * 128 + head * 16;
    #pragma unroll
    for (int i = 0; i < 16; ++i) { lg[i] = ca[i]; lg[16 + i] = ta[i]; }
    float mx = lg[0];
    #pragma unroll
    for (int i = 1; i < 32; ++i) mx = fmaxf(mx, lg[i]);
    float ssum = 0.f;
    #pragma unroll
    for (int i = 0; i < 32; ++i) { lg[i] = __expf(lg[i] - mx); ssum += lg[i]; }
    const float inv = 1.f / ssum;

    const float* co = coff  + (size_t)q * 256 + head * 32; // (l*4+p)*2+c
    const float* ts = tsoff + (size_t)q * 256 + head * 32; // ((l*2+tw)*2+tp)*2+c

    float4 accA = {0.f, 0.f, 0.f, 0.f};
    float4 accB = {0.f, 0.f, 0.f, 0.f};

    for (int l = 0; l < N_LEVELS; ++l) {
        const float rx = ref[((size_t)q * 4 + l) * 2 + 0];
        const float ry = ref[((size_t)q * 4 + l) * 2 + 1];
        const int   Hi = c_Hs[l], Wi = c_Ws[l];
        const float Hg = (float)Hi, Wg = (float)Wi;
        const float* vbase = value + (size_t)c_start[l] * 256 + head * 32 + d0;

        #pragma unroll
        for (int p = 0; p < 8; ++p) {
            float lx, ly, wgt;
            if (p < 4) {                    // current-frame points
                lx  = rx + co[(l * 4 + p) * 2 + 0] / Wg;
                ly  = ry + co[(l * 4 + p) * 2 + 1] / Hg;
                wgt = lg[l * 4 + p] * inv;
            } else {                        // temporal points
                const int j  = p - 4;
                const int tw = j >> 1;
                const int tp = j & 1;
                const float tx = toff[(((size_t)q * 4 + l) * 2 + tw) * 2 + 0];
                const float ty = toff[(((size_t)q * 4 + l) * 2 + tw) * 2 + 1];
                lx  = rx + tx + ts[((l * 2 + tw) * 2 + tp) * 2 + 0] / Wg;
                ly  = ry + ty + ts[((l * 2 + tw) * 2 + tp) * 2 + 1] / Hg;
                wgt = lg[16 + l * 4 + j] * inv;
            }
            // bilinear, align_corners=False, zero padding
            const float x  = lx * Wg - 0.5f;
            const float y  = ly * Hg - 0.5f;
            const float xf = floorf(x), yf = floorf(y);
            const int   x0 = (int)xf,   y0 = (int)yf;
            const float wx = x - xf,    wy = y - yf;

            #pragma unroll
            for (int dy = 0; dy < 2; ++dy) {
                const int yi = y0 + dy;
                if (yi < 0 || yi >= Hi) continue;
                const float wyc = dy ? wy : (1.f - wy);
                #pragma unroll
                for (int dx = 0; dx < 2; ++dx) {
                    const int xi = x0 + dx;
                    if (xi < 0 || xi >= Wi) continue;
                    const float cw = wgt * wyc * (dx ? wx : (1.f - wx));
                    const float4* vp = (const float4*)(vbase + (size_t)(yi * Wi + xi) * 256);
                    const float4 vA = vp[0];
                    const float4 vB = vp[1];
                    accA.x += cw * vA.x; accA.y += cw * vA.y;
                    accA.z += cw * vA.z; accA.w += cw * vA.w;
                    accB.x += cw * vB.x; accB.y += cw * vB.y;
                    accB.z += cw * vB.z; accB.w += cw * vB.w;
                }
            }
        }
    }

    float4* op = (float4*)(out + (size_t)q * 256 + head * 32 + d0);
    op[0] = accA;
    op[1] = accB;
}

// ---------------------------------------------------------------------------
// Host-side launcher
// ---------------------------------------------------------------------------
extern "C" void kernel_launch(void* const* d_in, const int* in_sizes, int n_in,
                              void* d_out, int out_size, void* d_ws, size_t ws_size,
                              hipStream_t stream)
{
    const float* query  = (const float*)d_in[0];
    const float* ref    = (const float*)d_in[1];
    const float* toff   = (const float*)d_in[2];
    const float* inflat = (const float*)d_in[3];
    // d_in[4], d_in[5] : spatial shapes / level starts (compile-time constants)
    const float* Wv  = (const float*)d_in[6];
    const float* bv  = (const float*)d_in[7];
    const float* Ws  = (const float*)d_in[8];
    const float* bs  = (const float*)d_in[9];
    const float* Wa  = (const float*)d_in[10];
    const float* ba  = (const float*)d_in[11];
    const float* Wts = (const float*)d_in[12];
    const float* bts = (const float*)d_in[13];
    const float* Wta = (const float*)d_in[14];
    const float* bta = (const float*)d_in[15];
    const float* Wo  = (const float*)d_in[16];
    const float* bo  = (const float*)d_in[17];

    float* wsp   = (float*)d_ws;
    float* value = wsp;                              // 8160*256
    float* coff  = value + (size_t)LEN_Q * 256;      // 8160*256
    float* cattn = coff  + (size_t)LEN_Q * 256;      // 8160*128
    float* tsoff = cattn + (size_t)LEN_Q * 128;      // 8160*256
    float* tattn = tsoff + (size_t)LEN_Q * 256;      // 8160*128
    float* aout  = tattn + (size_t)LEN_Q * 128;      // 8160*256

    const dim3 blk(256);
    const int  mgrid = (MTILES + 7) / 8;             // 64 row-blocks of 128
    const dim3 g256(4, mgrid);                       // N=256 -> 4 col-blocks
    const dim3 g128(2, mgrid);                       // N=128 -> 2 col-blocks

    gemm_f32_wmma<<<g256, blk, 0, stream>>>(inflat, Wv,  bv,  value, 256);
    gemm_f32_wmma<<<g256, blk, 0, stream>>>(query,  Ws,  bs,  coff,  256);
    gemm_f32_wmma<<<g128, blk, 0, stream>>>(query,  Wa,  ba,  cattn, 128);
    gemm_f32_wmma<<<g256, blk, 0, stream>>>(query,  Wts, bts, tsoff, 256);
    gemm_f32_wmma<<<g128, blk, 0, stream>>>(query,  Wta, bta, tattn, 128);

    msda_sample<<<dim3(LEN_Q / 8), dim3(256), 0, stream>>>(
        ref, toff, value, coff, cattn, tsoff, tattn, aout);

    gemm_f32_wmma<<<g256, blk, 0, stream>>>(aout, Wo, bo, (float*)d_out, 256);
}